// CornerDecoder_4346506903764
// MI455X (gfx1250) — compile-verified
//
#include <hip/hip_runtime.h>
#include <stdint.h>

typedef unsigned int u32;
typedef unsigned long long u64;

#define CC 80
#define HH 384
#define WW 384
#define HW (HH * WW)            // 147456
#define KTOP 100
#define NDET 1000
#define AE_THRESH 0.5f
#define TH_ROWS 8
#define NTILEY (HH / TH_ROWS)   // 48
#define NBINS 1024
#define CAND_CAP 4096
#define NPAIR_PAD 16384

// ---- workspace byte offsets ----
// 0      : hist_tl  (1024 u32)
// 4096   : hist_br  (1024 u32)
// 8192   : cnt_tl, 8196: cnt_br, 8200: thr_tl, 8204: thr_br
// 16384  : cand_tl (4096 u64), 49152: cand_br (4096 u64)
// 81920  : tl arrays, 86016: br arrays  (sc@0, cls@512, xf@1024, yf@1536, tag@2048)
// 90112  : pair keys (16384 u64)  -> total ~216 KB

__device__ __forceinline__ u32 fkey(float f) {
    u32 u = __float_as_uint(f);
    return (u & 0x80000000u) ? ~u : (u | 0x80000000u);
}

// CDNA5 async global->LDS load (ASYNCcnt path), 16B per lane
__device__ __forceinline__ void async_ld_b128(u32 lds_addr, u64 gaddr) {
    asm volatile("global_load_async_to_lds_b128 %0, %1, off"
                 :: "v"(lds_addr), "v"(gaddr) : "memory");
}
__device__ __forceinline__ void wait_async() {
    asm volatile("s_wait_asynccnt 0" ::: "memory");
}

__global__ void init_kernel(u32* ws32) {
    int t = blockIdx.x * blockDim.x + threadIdx.x;
    if (t < 2052) ws32[t] = 0;   // hists + counters + thresholds
}

// Fused sigmoid-free NMS scan. mode 0: histogram of local-max keys.
// mode 1: compact candidates with key >= *thrPtr.
__global__ __launch_bounds__(256)
void nms_scan_kernel(const float* __restrict__ heat, u32* __restrict__ hist,
                     const u32* __restrict__ thrPtr, u32* __restrict__ cnt,
                     u64* __restrict__ cand, int mode)
{
    __shared__ __align__(16) float tile[TH_ROWS + 2][WW];
    __shared__ u32 lhist[NBINS];
    const int tid = threadIdx.x;
    const int c   = blockIdx.x / NTILEY;
    const int ty  = blockIdx.x % NTILEY;
    const int y0  = ty * TH_ROWS;
    const float NEG = __int_as_float(0xff800000);

    if (mode == 0)
        for (int b = tid; b < NBINS; b += 256) lhist[b] = 0;
    const u32 thr = (mode == 1) ? *thrPtr : 0u;

    // async-load (TH_ROWS+2) x 384 float tile, 96 x 16B chunks per row
    const u32 lds_base = (u32)(uintptr_t)(&tile[0][0]);
    const float* chanBase = heat + (size_t)c * HW;
    for (int i = tid; i < (TH_ROWS + 2) * 96; i += 256) {
        int r = i / 96, k = i % 96;
        int gy = y0 - 1 + r;
        if (gy >= 0 && gy < HH)
            async_ld_b128(lds_base + (u32)(r * WW * 4 + k * 16),
                          (u64)(uintptr_t)(chanBase + (size_t)gy * WW + k * 4));
    }
    if (y0 == 0)
        for (int x = tid; x < WW; x += 256) tile[0][x] = NEG;
    if (y0 + TH_ROWS == HH)
        for (int x = tid; x < WW; x += 256) tile[TH_ROWS + 1][x] = NEG;
    wait_async();
    __syncthreads();

    for (int e = tid; e < TH_ROWS * WW; e += 256) {
        int ly = e / WW, x = e % WW, r = ly + 1;
        float v  = tile[r][x];
        float l  = (x > 0)      ? tile[r][x - 1]     : NEG;
        float rr = (x < WW - 1) ? tile[r][x + 1]     : NEG;
        float u0 = tile[r - 1][x];
        float ul = (x > 0)      ? tile[r - 1][x - 1] : NEG;
        float ur = (x < WW - 1) ? tile[r - 1][x + 1] : NEG;
        float d0 = tile[r + 1][x];
        float dl = (x > 0)      ? tile[r + 1][x - 1] : NEG;
        float dr = (x < WW - 1) ? tile[r + 1][x + 1] : NEG;
        float m = fmaxf(fmaxf(fmaxf(l, rr), fmaxf(u0, d0)),
                        fmaxf(fmaxf(ul, ur), fmaxf(dl, dr)));
        if (v >= m) {   // local maximum (sigmoid is monotonic -> logit space OK)
            u32 key = fkey(v);
            if (mode == 0) {
                atomicAdd(&lhist[key >> 22], 1u);
            } else if (key >= thr) {
                u32 pos = atomicAdd(cnt, 1u);
                if (pos < CAND_CAP) {
                    u32 idx = (u32)(c * HW + (y0 + ly) * WW + x);
                    cand[pos] = ((u64)key << 32) | (u64)(0xFFFFFFFFu - idx);
                }
            }
        }
    }
    if (mode == 0) {
        __syncthreads();
        for (int b = tid; b < NBINS; b += 256)
            if (lhist[b]) atomicAdd(&hist[b], lhist[b]);
    }
}

__global__ void thresh_kernel(u32* ws32) {
    int t = threadIdx.x;
    if (t < 2) {
        const u32* hist = ws32 + (t ? 1024 : 0);
        u32 cum = 0, key = 0;
        for (int b = NBINS - 1; b >= 0; --b) {
            cum += hist[b];
            if (cum >= KTOP) { key = (u32)b << 22; break; }
        }
        ws32[2050 + t] = key;
    }
}

// Single-block bitonic top-100 + gathers for one corner type.
__global__ __launch_bounds__(1024)
void select_kernel(const float* __restrict__ heat, const float* __restrict__ embd,
                   const float* __restrict__ offs,
                   const u64* __restrict__ cand, const u32* __restrict__ cnt,
                   float* __restrict__ arr)
{
    __shared__ u64 s[CAND_CAP];
    int tid = threadIdx.x;
    u32 n = *cnt; if (n > CAND_CAP) n = CAND_CAP;
    for (int i = tid; i < CAND_CAP; i += 1024) s[i] = (i < (int)n) ? cand[i] : 0ull;
    __syncthreads();
    for (int k = 2; k <= CAND_CAP; k <<= 1)
        for (int j = k >> 1; j > 0; j >>= 1) {
            for (int t = tid; t < CAND_CAP; t += 1024) {
                int p = t ^ j;
                if (p > t) {
                    u64 a = s[t], b = s[p];
                    bool sw = ((t & k) == 0) ? (a < b) : (a > b);  // descending
                    if (sw) { s[t] = b; s[p] = a; }
                }
            }
            __syncthreads();
        }
    if (tid < KTOP) {
        u64 e = s[tid];
        u32 idx = 0xFFFFFFFFu - (u32)(e & 0xFFFFFFFFull);
        if (e == 0ull || idx >= (u32)(CC * HW)) idx = 0;
        int cc  = (int)(idx / HW);
        int rem = (int)(idx % HW);
        int y = rem / WW, x = rem % WW;
        float logit = heat[idx];
        float sc = 1.0f / (1.0f + __expf(-logit));
        arr[tid] = sc;
        ((int*)((char*)arr + 512))[tid]    = cc;
        ((float*)((char*)arr + 1024))[tid] = (float)x + offs[y * WW + x];
        ((float*)((char*)arr + 1536))[tid] = (float)y + offs[HW + y * WW + x];
        ((float*)((char*)arr + 2048))[tid] = embd[y * WW + x];
    }
}

__device__ __forceinline__ float pair_score(const float* tlA, const float* brA,
                                            int i, int j) {
    float sc = 0.5f * (tlA[i] + brA[j]);
    int   ci = ((const int*)((const char*)tlA + 512))[i];
    int   cj = ((const int*)((const char*)brA + 512))[j];
    float xi = ((const float*)((const char*)tlA + 1024))[i];
    float xj = ((const float*)((const char*)brA + 1024))[j];
    float yi = ((const float*)((const char*)tlA + 1536))[i];
    float yj = ((const float*)((const char*)brA + 1536))[j];
    float ti = ((const float*)((const char*)tlA + 2048))[i];
    float tj = ((const float*)((const char*)brA + 2048))[j];
    bool invalid = (fabsf(ti - tj) > AE_THRESH) || (ci != cj) || (xi > xj) || (yi > yj);
    return invalid ? -1.0f : sc;
}

__global__ __launch_bounds__(256)
void pairs_kernel(const float* __restrict__ tlA, const float* __restrict__ brA,
                  u64* __restrict__ pair)
{
    int p = blockIdx.x * 256 + threadIdx.x;
    if (p >= NPAIR_PAD) return;
    if (p < KTOP * KTOP) {
        float sc = pair_score(tlA, brA, p / KTOP, p % KTOP);
        pair[p] = ((u64)fkey(sc) << 32) | (u64)(0xFFFFFFFFu - (u32)p);
    } else {
        pair[p] = 0ull;
    }
}

// Single-block bitonic sort (16384 u64, workspace-resident) + emit detections.
__global__ __launch_bounds__(1024)
void sort_emit_kernel(u64* __restrict__ pair, const float* __restrict__ tlA,
                      const float* __restrict__ brA, float* __restrict__ out)
{
    int tid = threadIdx.x;
    for (int k = 2; k <= NPAIR_PAD; k <<= 1)
        for (int j = k >> 1; j > 0; j >>= 1) {
            for (int t = tid; t < NPAIR_PAD; t += 1024) {
                int p = t ^ j;
                if (p > t) {
                    u64 a = pair[t], b = pair[p];
                    bool sw = ((t & k) == 0) ? (a < b) : (a > b);
                    if (sw) { pair[t] = b; pair[p] = a; }
                }
            }
            __syncthreads();
        }
    if (tid < NDET) {
        u64 e = pair[tid];
        u32 p = 0xFFFFFFFFu - (u32)(e & 0xFFFFFFFFull);
        if (p >= (u32)(KTOP * KTOP)) p = 0;
        int i = (int)(p / KTOP), j = (int)(p % KTOP);
        float sc = pair_score(tlA, brA, i, j);
        out[tid * 8 + 0] = ((const float*)((const char*)tlA + 1024))[i];
        out[tid * 8 + 1] = ((const float*)((const char*)tlA + 1536))[i];
        out[tid * 8 + 2] = ((const float*)((const char*)brA + 1024))[j];
        out[tid * 8 + 3] = ((const float*)((const char*)brA + 1536))[j];
        out[tid * 8 + 4] = sc;
        out[tid * 8 + 5] = tlA[i];
        out[tid * 8 + 6] = brA[j];
        out[tid * 8 + 7] = (float)(((const int*)((const char*)tlA + 512))[i]);
    }
}

extern "C" void kernel_launch(void* const* d_in, const int* in_sizes, int n_in,
                              void* d_out, int out_size, void* d_ws, size_t ws_size,
                              hipStream_t stream) {
    (void)in_sizes; (void)n_in; (void)out_size; (void)ws_size;
    const float* tl_heat = (const float*)d_in[0];
    const float* br_heat = (const float*)d_in[1];
    const float* tl_embd = (const float*)d_in[2];
    const float* br_embd = (const float*)d_in[3];
    const float* tl_offs = (const float*)d_in[4];
    const float* br_offs = (const float*)d_in[5];
    char* ws  = (char*)d_ws;
    u32* ws32 = (u32*)d_ws;
    u32* hist_tl = (u32*)(ws + 0);
    u32* hist_br = (u32*)(ws + 4096);
    u32* cnt_tl  = (u32*)(ws + 8192);
    u32* cnt_br  = (u32*)(ws + 8196);
    u32* thr_tl  = (u32*)(ws + 8200);
    u32* thr_br  = (u32*)(ws + 8204);
    u64* cand_tl = (u64*)(ws + 16384);
    u64* cand_br = (u64*)(ws + 49152);
    float* tlA   = (float*)(ws + 81920);
    float* brA   = (float*)(ws + 86016);
    u64* pair    = (u64*)(ws + 90112);
    float* out   = (float*)d_out;

    const int nblk = CC * NTILEY;  // 3840
    init_kernel<<<9, 256, 0, stream>>>(ws32);
    nms_scan_kernel<<<nblk, 256, 0, stream>>>(tl_heat, hist_tl, thr_tl, cnt_tl, cand_tl, 0);
    nms_scan_kernel<<<nblk, 256, 0, stream>>>(br_heat, hist_br, thr_br, cnt_br, cand_br, 0);
    thresh_kernel<<<1, 32, 0, stream>>>(ws32);
    nms_scan_kernel<<<nblk, 256, 0, stream>>>(tl_heat, hist_tl, thr_tl, cnt_tl, cand_tl, 1);
    nms_scan_kernel<<<nblk, 256, 0, stream>>>(br_heat, hist_br, thr_br, cnt_br, cand_br, 1);
    select_kernel<<<1, 1024, 0, stream>>>(tl_heat, tl_embd, tl_offs, cand_tl, cnt_tl, tlA);
    select_kernel<<<1, 1024, 0, stream>>>(br_heat, br_embd, br_offs, cand_br, cnt_br, brA);
    pairs_kernel<<<NPAIR_PAD / 256, 256, 0, stream>>>(tlA, brA, pair);
    sort_emit_kernel<<<1, 1024, 0, stream>>>(pair, tlA, brA, out);
}